// GLSLayer_27032524161194
// MI455X (gfx1250) — compile-verified
//
#include <hip/hip_runtime.h>
#include <hip/hip_bf16.h>

// ---------------------------------------------------------------------------
// GLS chaotic-firing + cosine-logits for MI455X (gfx1250, wave32, WMMA)
//
//   Phase 1 (VALU-bound, dominant): per-element skew-tent first-passage
//            count k (B x N); IEEE f32 division, wave32 __any early exit.
//   Phase 2: row inverse norms of K and M (LDS tree reduction).
//   Phase 3: logits = (K @ M^T) * invKn * invMn on the matrix pipe via
//            V_WMMA_F32_16X16X4_F32 (f32 exact for k up to 4097).
// ---------------------------------------------------------------------------

typedef float v2f __attribute__((ext_vector_type(2)));
typedef float v8f __attribute__((ext_vector_type(8)));

static constexpr int Bn = 256;    // batch rows
static constexpr int Nn = 8192;   // neurons (K dim of GEMM)
static constexpr int Cn = 100;    // classes
static constexpr int MAX_ITERS = 4096;

// ---------------------------------------------------------------------------
// Phase 1: chaotic firing. One thread per (row, col) element.
//   num = (m<b) ? m : (1-m);  den = (m<b) ? b : (1-b);  m = num/den (IEEE)
//   k counts 1 + iterations while m stayed outside [x-e, x+e].
// ---------------------------------------------------------------------------
__global__ __launch_bounds__(256)
void gls_fire_kernel(const float* __restrict__ x, const float* __restrict__ bv,
                     const float* __restrict__ qv, const float* __restrict__ ev,
                     float* __restrict__ Kout)
{
    const int col = blockIdx.x * blockDim.x + threadIdx.x;   // coalesced
    const int row = blockIdx.y;

    const float bb = bv[col];
    float m        = qv[col];
    const float ee = ev[col];
    const float xx = x[(size_t)row * Nn + col];
    const float lo = xx - ee;
    const float hi = xx + ee;

    float k  = 1.0f;
    bool act = (m != 0.0f);

    for (int t = 0; t < MAX_ITERS; ++t) {
        if (!__any(act)) break;                 // whole wave32 converged
        const float num = (m < bb) ? m  : (1.0f - m);
        const float den = (m < bb) ? bb : (1.0f - bb);
        m = num / den;                          // IEEE f32 divide (chaotic map)
        const bool inval = (m < lo) || (m > hi);
        act = act && inval;
        k += act ? 1.0f : 0.0f;
    }
    Kout[(size_t)row * Nn + col] = k;
}

// ---------------------------------------------------------------------------
// Phase 2: inv[row] = 1 / max(||A[row,:]||, 1e-8)
// ---------------------------------------------------------------------------
__global__ __launch_bounds__(256)
void row_inv_norm_kernel(const float* __restrict__ A, float* __restrict__ inv,
                         int cols)
{
    __shared__ float s[256];
    const int row = blockIdx.x;
    const float* p = A + (size_t)row * cols;

    float acc = 0.0f;
    for (int j = threadIdx.x; j < cols; j += 256) {
        const float v = p[j];
        acc = fmaf(v, v, acc);
    }
    s[threadIdx.x] = acc;
    __syncthreads();
    for (int off = 128; off > 0; off >>= 1) {
        if (threadIdx.x < off) s[threadIdx.x] += s[threadIdx.x + off];
        __syncthreads();
    }
    if (threadIdx.x == 0) {
        const float n = fmaxf(sqrtf(s[0]), 1e-8f);
        inv[row] = 1.0f / n;
    }
}

// ---------------------------------------------------------------------------
// Phase 3: one wave32 per 16x32 output tile (two 16x16 WMMA accumulators
// sharing one A fragment).  Invalid (padded) B columns are clamped to row 0:
// garbage there only reaches D columns >= Cn, which are never stored, so the
// K-loop body is branch-free (2x global_load_b64 per accumulator + v_wmma),
// and EXEC stays all-1s as WMMA requires.
//
// A (16x4) layout: float2 per lane at K[tM*16 + (lane&15)][k + 2*(lane>>4)]
// B (4x16) layout: float2 per lane at M[tN*16 + (lane&15)][k + 2*(lane>>4)]
// C/D: VGPR v -> row (v + 8*(lane>>4)), col = lane&15.
// ---------------------------------------------------------------------------
__global__ __launch_bounds__(32)
void gls_logits_wmma_kernel(const float* __restrict__ Kmat,
                            const float* __restrict__ Mmat,
                            const float* __restrict__ invKn,
                            const float* __restrict__ invMn,
                            float* __restrict__ out)
{
    const int lane  = threadIdx.x;           // 0..31
    const int tM    = blockIdx.x;            // 0..15  (row tiles of K)
    const int tN2   = blockIdx.y;            // 0..3   (pairs of 16-col tiles)
    const int l15   = lane & 15;
    const int khalf = (lane >> 4) * 2;       // 0 or 2

    const int rowA  = tM * 16 + l15;         // 0..255
    const int colB0 = tN2 * 32 + l15;        // first  16-col tile
    const int colB1 = colB0 + 16;            // second 16-col tile
    const int cB0   = (colB0 < Cn) ? colB0 : 0;   // clamp: see header comment
    const int cB1   = (colB1 < Cn) ? colB1 : 0;

    const float* Arow  = Kmat + (size_t)rowA * Nn + khalf;
    const float* Brow0 = Mmat + (size_t)cB0 * Nn + khalf;
    const float* Brow1 = Mmat + (size_t)cB1 * Nn + khalf;

    v8f acc0 = {};
    v8f acc1 = {};
    #pragma unroll 4
    for (int k = 0; k < Nn; k += 4) {
        const v2f a  = *(const v2f*)(Arow  + k);
        const v2f b0 = *(const v2f*)(Brow0 + k);
        const v2f b1 = *(const v2f*)(Brow1 + k);
        acc0 = __builtin_amdgcn_wmma_f32_16x16x4_f32(
            false, a, false, b0, (short)0, acc0, false, false);
        acc1 = __builtin_amdgcn_wmma_f32_16x16x4_f32(
            false, a, false, b1, (short)0, acc1, false, false);
    }

    const float sN0 = (colB0 < Cn) ? invMn[colB0] : 0.0f;
    const float sN1 = (colB1 < Cn) ? invMn[colB1] : 0.0f;
    #pragma unroll
    for (int v = 0; v < 8; ++v) {
        const int r = tM * 16 + v + (lane >> 4) * 8;
        const float sK = invKn[r];
        if (colB0 < Cn) out[(size_t)r * Cn + colB0] = acc0[v] * sK * sN0;
        if (colB1 < Cn) out[(size_t)r * Cn + colB1] = acc1[v] * sK * sN1;
    }
}

// ---------------------------------------------------------------------------
// Launcher (graph-capture safe).  Inputs: x(B,N), b(N), q(N), e(N), M(C,N),
// all f32.  Output: logits (B,C) f32.  Workspace: K + inverse norms (~8.4MB).
// ---------------------------------------------------------------------------
extern "C" void kernel_launch(void* const* d_in, const int* in_sizes, int n_in,
                              void* d_out, int out_size, void* d_ws, size_t ws_size,
                              hipStream_t stream) {
    const float* x = (const float*)d_in[0];
    const float* b = (const float*)d_in[1];
    const float* q = (const float*)d_in[2];
    const float* e = (const float*)d_in[3];
    const float* M = (const float*)d_in[4];
    float* out = (float*)d_out;

    float* Kmat  = (float*)d_ws;                 // B*N floats = 8 MB
    float* invKn = Kmat + (size_t)Bn * Nn;       // B floats
    float* invMn = invKn + 256;                  // C floats (padded region)

    // Phase 1: chaotic firing -> K (B x N)
    gls_fire_kernel<<<dim3(Nn / 256, Bn), 256, 0, stream>>>(x, b, q, e, Kmat);

    // Phase 2: inverse norms
    row_inv_norm_kernel<<<Bn, 256, 0, stream>>>(Kmat, invKn, Nn);
    row_inv_norm_kernel<<<Cn, 256, 0, stream>>>(M,    invMn, Nn);

    // Phase 3: WMMA f32 GEMM + cosine scaling -> out (B x C)
    gls_logits_wmma_kernel<<<dim3(Bn / 16, 4), 32, 0, stream>>>(
        Kmat, M, invKn, invMn, out);
}